// RollingStatsLayer_83811991814672
// MI455X (gfx1250) — compile-verified
//
#include <hip/hip_runtime.h>
#include <math.h>

typedef __attribute__((ext_vector_type(2))) float v2f;
typedef __attribute__((ext_vector_type(8))) float v8f;

#define T_IN   262144
#define NF     32
#define WIN    64
#define T_OUT  (T_IN - WIN + 1)        // 262081
#define OUT_STRIDE (NF * 5)            // 160

// ---------------------------------------------------------------------------
// Kernel 1: rolling sum + sum-of-squares via fp32 WMMA (exact fp32 math).
// One wave computes a 16(time) x 16(feature) output tile:
//   S = A(16x80, banded ones) * X(80x16)  as 20 chained 16x16x4 WMMAs (x2 for
//   sum of squares). Clamped row indices make all loads unconditional; rows
//   past T_IN only feed A-entries belonging to unwritten outputs.
// ---------------------------------------------------------------------------
__global__ __launch_bounds__(32)
void rolling_sum_wmma(const float* __restrict__ x, float* __restrict__ out)
{
    const int t0   = blockIdx.x * 16;      // first output row of this tile
    const int f0   = blockIdx.y * 16;      // first feature column
    const int lane = threadIdx.x;          // 0..31 (wave32)
    const int n    = lane & 15;            // B column / A row index
    const int half = lane >> 4;            // lane half -> K pair select
    const int col  = f0 + n;

    // ---- Load all 40 B elements (20 chunks x 2 rows) unconditionally ----
    float b0[20], b1[20];
    if (t0 + 79 < T_IN) {
        // Fast path: pure immediate offsets (512*j and 512*j+128 bytes).
        const float* p = x + (size_t)(t0 + 2 * half) * NF + col;
#pragma unroll
        for (int j = 0; j < 20; ++j) {
            b0[j] = p[j * 4 * NF];
            b1[j] = p[j * 4 * NF + NF];
        }
    } else {
        // Last tile: clamp rows into range (contributions of clamped rows
        // only reach output rows t >= T_OUT, which are never stored).
#pragma unroll
        for (int j = 0; j < 20; ++j) {
            int r0 = t0 + 4 * j + 2 * half;
            int r1 = r0 + 1;
            r0 = (r0 < T_IN - 1) ? r0 : (T_IN - 1);
            r1 = (r1 < T_IN - 1) ? r1 : (T_IN - 1);
            b0[j] = x[(size_t)r0 * NF + col];
            b1[j] = x[(size_t)r1 * NF + col];
        }
    }

    // ---- 40 chained fp32 WMMAs ----
    v8f acc_s = {};
    v8f acc_q = {};
#pragma unroll
    for (int j = 0; j < 20; ++j) {
        const int s0 = 4 * j + 2 * half;   // K index of this lane's VGPR0 slot
        const int s1 = s0 + 1;

        v2f B; B.x = b0[j];          B.y = b1[j];
        v2f Q; Q.x = b0[j] * b0[j];  Q.y = b1[j] * b1[j];

        // Banded-ones A: A[m,s] = 1 iff 0 <= s-m <= 63   (m = n here)
        v2f A;
        A.x = ((unsigned)(s0 - n) <= 63u) ? 1.0f : 0.0f;
        A.y = ((unsigned)(s1 - n) <= 63u) ? 1.0f : 0.0f;

        acc_s = __builtin_amdgcn_wmma_f32_16x16x4_f32(
                    false, A, false, B, (short)0, acc_s, false, false);
        acc_q = __builtin_amdgcn_wmma_f32_16x16x4_f32(
                    false, A, false, Q, (short)0, acc_q, false, false);
    }

    // ---- Epilogue: mean / std / sum, non-temporal stores ----
#pragma unroll
    for (int v = 0; v < 8; ++v) {
        const int m = v + 8 * half;
        const int t = t0 + m;
        if (t < T_OUT) {
            const float S    = acc_s[v];
            const float S2   = acc_q[v];
            const float mean = S * (1.0f / 64.0f);
            const float var  = S2 * (1.0f / 64.0f) - mean * mean;
            const float sd   = sqrtf(fmaxf(var, 0.0f));
            float* row = out + (size_t)t * OUT_STRIDE;
            __builtin_nontemporal_store(mean, row + 0 * NF + col);  // mean
            __builtin_nontemporal_store(sd,   row + 1 * NF + col);  // std
            __builtin_nontemporal_store(S,    row + 4 * NF + col);  // sum
        }
    }
}

// ---------------------------------------------------------------------------
// Kernel 2: rolling min/max via van Herk / Gil-Werman.
// 256 threads = 32 feature columns x 8 time-blocks of 64 rows.
// Phase 1: thread (f, j) forward-scans block (g*7 + j), prefix min/max -> LDS.
// Phase 2 (j < 7): backward suffix scan in registers;
//   out[64b + i] = op(sfx[i], pfx_{b+1}[i-1])  (i>0), out[64b] = block reduce.
// ---------------------------------------------------------------------------
#define NBP 8   // prefix blocks per workgroup
#define NBO 7   // output blocks per workgroup

__global__ __launch_bounds__(256)
void rolling_minmax(const float* __restrict__ x, float* __restrict__ out)
{
    extern __shared__ float lds[];                   // 2 * NBP * 64 * 32 floats
    float* pmin = lds;
    float* pmax = lds + NBP * 64 * NF;

    const int f  = threadIdx.x & 31;                 // feature column
    const int j  = threadIdx.x >> 5;                 // block slot 0..7 (wave-uniform)
    const long long bb   = (long long)blockIdx.x * NBO + j;  // prefix block id
    const long long base = bb * 64;

    // ---- Phase 1: prefix min/max of block bb into LDS (clamped reads) ----
    // Entries containing clamped rows are only consumed for windows whose end
    // row is valid, so clamping never pollutes a consumed prefix value.
    float rmin =  INFINITY;
    float rmax = -INFINITY;
#pragma unroll 8
    for (int i = 0; i < 64; ++i) {
        long long t = base + i;
        t = (t < T_IN - 1) ? t : (T_IN - 1);
        const float v = x[t * NF + f];
        rmin = fminf(rmin, v);
        rmax = fmaxf(rmax, v);
        pmin[(j * 64 + i) * NF + f] = rmin;
        pmax[(j * 64 + i) * NF + f] = rmax;
    }
    __syncthreads();

    // ---- Phase 2: suffix scan + combine for output block bb (j < NBO) ----
    // base < T_OUT  =>  bb <= 4095  =>  all suffix reads (<= 64*bb+63) valid.
    if (j < NBO && base < T_OUT) {
        float smin =  INFINITY;
        float smax = -INFINITY;
#pragma unroll 8
        for (int i = 63; i >= 0; --i) {
            const long long t = base + i;            // candidate output row
            const float v = x[t * NF + f];
            smin = fminf(smin, v);
            smax = fmaxf(smax, v);
            if (t < T_OUT) {
                float omin, omax;
                if (i == 0) {                        // window == block bb
                    omin = smin; omax = smax;
                } else {                             // spans bb and bb+1
                    omin = fminf(smin, pmin[((j + 1) * 64 + (i - 1)) * NF + f]);
                    omax = fmaxf(smax, pmax[((j + 1) * 64 + (i - 1)) * NF + f]);
                }
                float* row = out + (size_t)t * OUT_STRIDE;
                __builtin_nontemporal_store(omin, row + 2 * NF + f);  // min
                __builtin_nontemporal_store(omax, row + 3 * NF + f);  // max
            }
        }
    }
}

// ---------------------------------------------------------------------------
extern "C" void kernel_launch(void* const* d_in, const int* in_sizes, int n_in,
                              void* d_out, int out_size, void* d_ws, size_t ws_size,
                              hipStream_t stream)
{
    (void)in_sizes; (void)n_in; (void)out_size; (void)d_ws; (void)ws_size;
    const float* x   = (const float*)d_in[0];
    float*       out = (float*)d_out;

    // Kernel 1: mean / std / sum via fp32 WMMA.
    dim3 g1((T_OUT + 15) / 16, NF / 16);             // 16381 x 2 tiles
    rolling_sum_wmma<<<g1, 32, 0, stream>>>(x, out);

    // Kernel 2: min / max via van Herk two-scan.
    const int nblk = (T_OUT + 63) / 64;              // 4096 output blocks
    const int nwg  = (nblk + NBO - 1) / NBO;         // 586 workgroups
    const size_t ldsBytes = (size_t)2 * NBP * 64 * NF * sizeof(float);  // 128 KB
    rolling_minmax<<<nwg, 256, ldsBytes, stream>>>(x, out);
}